// SpecialSpmmFinal_rel_31430570672194
// MI455X (gfx1250) — compile-verified
//
#include <hip/hip_runtime.h>

// segment_sum(edge_w[E,32], edge[0], N1) -> out[N1,32]
// Memory-bound scatter-add. F == 32 == wave32 width.

typedef float f4 __attribute__((ext_vector_type(4)));

#define F_ 32

__global__ void zero_out_kernel(f4* __restrict__ out, int n4) {
  int i = blockIdx.x * blockDim.x + threadIdx.x;
  const int stride = gridDim.x * blockDim.x;
  f4 z = {0.0f, 0.0f, 0.0f, 0.0f};
  for (; i < n4; i += stride) {
    out[i] = z;
  }
}

// One wave processes 4 edges per iteration:
//   lane = 8*group + sub ; group in [0,4) selects the edge,
//   sub in [0,8) selects the float4 chunk of the 32-float row.
// Wave-level access pattern: one 512B contiguous global_load_b128 burst of
// edge_w (non-temporal: streamed once, keep L2 for the atomics' working set),
// then 4 global_atomic_add_f32 per lane into the 12.8MB output that lives in L2.
__global__ void scatter_add_kernel(const int* __restrict__ src,
                                   const float* __restrict__ edge_w,
                                   float* __restrict__ out,
                                   int E) {
  const int lane  = threadIdx.x & 31;
  const int group = lane >> 3;   // which of the 4 edges this wave covers
  const int sub   = lane & 7;    // which float4 chunk of the row
  const int wave  = (blockIdx.x * blockDim.x + threadIdx.x) >> 5;
  const int nwave = (gridDim.x * blockDim.x) >> 5;
  const int estep = nwave * 4;

  for (int e0 = wave * 4; e0 < E; e0 += estep) {
    const int e = e0 + group;
    if (e < E) {
      // Prefetch the next chunk this wave will touch (global_prefetch_b8).
      const int en = e + estep;
      if (en < E) {
        __builtin_prefetch(edge_w + (size_t)en * F_ + sub * 4, 0, 1);
      }

      const int idx = src[e];  // 4 distinct addrs per wave; broadcast within groups
      const f4* wp = (const f4*)(edge_w + (size_t)e * F_) + sub;
      f4 w = __builtin_nontemporal_load(wp);  // TH=NT: stream, don't pollute L2

      float* o = out + (size_t)idx * F_ + sub * 4;
      atomicAdd(o + 0, w.x);   // non-returning global_atomic_add_f32 (L2-resident)
      atomicAdd(o + 1, w.y);
      atomicAdd(o + 2, w.z);
      atomicAdd(o + 3, w.w);
    }
  }
}

extern "C" void kernel_launch(void* const* d_in, const int* in_sizes, int n_in,
                              void* d_out, int out_size, void* d_ws, size_t ws_size,
                              hipStream_t stream) {
  // setup_inputs order: edge [2,E] int32, edge_w [E,32] f32, then scalars.
  const int*   edge   = (const int*)d_in[0];
  const float* edge_w = (const float*)d_in[1];
  const int E = in_sizes[0] / 2;          // edge is [2, E]
  const int* src = edge;                  // edge[0] = first E entries

  float* out = (float*)d_out;             // [N1, 32]; out_size = N1*32

  // 1) Zero the output (harness poisons d_out; re-zero every call => deterministic).
  {
    const int n4 = out_size / 4;          // 32 floats/row -> always divisible by 4
    const int threads = 256;
    int blocks = (n4 + threads * 4 - 1) / (threads * 4);
    if (blocks > 2048) blocks = 2048;
    if (blocks < 1)    blocks = 1;
    zero_out_kernel<<<blocks, threads, 0, stream>>>((f4*)out, n4);
  }

  // 2) Scatter-add. 3072 blocks x 256 threads = 24576 wave32s: enough
  //    memory-level parallelism to saturate 23.3 TB/s while grid-striding.
  {
    const int threads = 256;
    int blocks = 3072;
    const int waves_needed = (E + 3) / 4;                // one wave per 4 edges
    const int blocks_needed = (waves_needed * 32 + threads - 1) / threads;
    if (blocks > blocks_needed) blocks = blocks_needed;
    if (blocks < 1) blocks = 1;
    scatter_add_kernel<<<blocks, threads, 0, stream>>>(src, edge_w, out, E);
  }
}